// MyModel_87522843560546
// MI455X (gfx1250) — compile-verified
//
#include <hip/hip_runtime.h>
#include <hip/hip_bf16.h>

// Masked GRU, B=4096, T=4096, F=1, UNITS=1.
// One wave32 per block, lane = batch row. x tiles (32 rows x 128 steps) are
// DMA'd HBM->LDS by the CDNA5 Tensor Data Mover (tensor_load_to_lds, TENSORcnt),
// one descriptor per tile, double-buffered. TDM hardware padding (4 DWORDs per
// 128-DWORD row) gives a 132-float LDS row stride -> bank-conflict-free reads.
// All exp() log2e scalings are folded into the weights so the serial recurrence
// hits v_exp_f32 directly (shortest possible dependent chain per timestep).

#define B_DIM 4096
#define T_DIM 4096
#define ROWS 32          // rows (batch elements) per block == wave32 lanes
#define CHUNK 128        // timesteps per tile
#define NCHUNK (T_DIM / CHUNK)
#define LDS_STRIDE 132   // floats per row in LDS: 128 + 4 pad (TDM pad_amount)

typedef unsigned int u32;
typedef __attribute__((ext_vector_type(4))) unsigned int v4u;
typedef __attribute__((ext_vector_type(8))) unsigned int v8u;

// Issue one TDM load: 2D tile, 32 rows x 128 fp32, row stride T_DIM elements,
// into LDS at lds_addr with 4-DWORD padding after every 128 DWORDs.
__device__ __forceinline__ void tdm_load_tile(const float* gptr, u32 lds_addr) {
    const unsigned long long ga = (unsigned long long)(uintptr_t)gptr;

    v4u g0;
    g0.x = 1u;                                   // count=1 (valid user descriptor)
    g0.y = lds_addr;                             // lds_addr[31:0] (bytes)
    g0.z = (u32)(ga & 0xFFFFFFFFu);              // global_addr[31:0]
    g0.w = (u32)((ga >> 32) & 0x01FFFFFFu)       // global_addr[56:32]
         | 0x80000000u;                          // type=2 ("image") at bits 127:126

    v8u g1;
    g1.s0 = (2u << 16)                           // data_size = 4 bytes
          | (1u << 20)                           // pad_enable
          | (6u << 22)                           // pad_interval: 128 DWORDs
          | (3u << 25);                          // pad_amount: 4 DWORDs
    g1.s1 = (u32)(T_DIM & 0xFFFF) << 16;         // tensor_dim0[15:0] @ bits 63:48
    g1.s2 = (u32)(T_DIM >> 16)                   // tensor_dim0[31:16]
          | ((u32)ROWS << 16);                   // tensor_dim1[15:0] @ bits 111:80
    g1.s3 = ((u32)CHUNK << 16);                  // tile_dim0 = 128 @ bits 127:112
    g1.s4 = (u32)ROWS;                           // tile_dim1 = 32 (tile_dim2 = 0)
    g1.s5 = (u32)T_DIM;                          // tensor_dim0_stride[31:0] = 4096
    g1.s6 = 0u;                                  // stride[47:32]=0, dim1_stride=0
    g1.s7 = 0u;

    // 2D tensor: VADDR2/VADDR3 omitted (NULL). Tracked with TENSORcnt.
    asm volatile("tensor_load_to_lds %0, %1" :: "s"(g0), "s"(g1) : "memory");
}

__global__ __launch_bounds__(ROWS) void gru_masked_tdm_kernel(
        const float* __restrict__ x,
        const float* __restrict__ kernel,      // (1, 3)
        const float* __restrict__ rkernel,     // (1, 3)
        const float* __restrict__ bias,        // (2, 3)
        float* __restrict__ hout)              // (B, 1)
{
    __shared__ __align__(16) float tile[2][ROWS * LDS_STRIDE];

    const int lane = threadIdx.x;                 // 0..31
    const int row0 = blockIdx.x * ROWS;
    const float* xrow0 = x + (size_t)row0 * T_DIM;  // wave-uniform tile base

    const u32 lds_base[2] = { (u32)(uintptr_t)&tile[0][0],
                              (u32)(uintptr_t)&tile[1][0] };

    // Fold exp->exp2 scaling into the (uniform, scalar) weights:
    //   sigmoid(v) = rcp(1 + exp2(-log2e * v))
    //   tanh(v)    = (1-e)*rcp(1+e), e = exp2(-2*log2e * v)
    const float NL2E  = -1.4426950408889634f;     // -log2(e)
    const float N2L2E = -2.8853900817779268f;     // -2*log2(e)
    const float k0 = NL2E  * kernel[0], k1 = NL2E  * kernel[1], k2 = N2L2E * kernel[2];
    const float r0 = NL2E  * rkernel[0], r1 = NL2E * rkernel[1], r2 = N2L2E * rkernel[2];
    const float bz = NL2E  * bias[0], br = NL2E  * bias[1], bh = N2L2E * bias[2];
    const float cz = NL2E  * bias[3], cr = NL2E  * bias[4], ch = N2L2E * bias[5];

    // Prime the double buffer: tiles 0 and 1 in flight (TENSORcnt = 2).
    tdm_load_tile(xrow0 + 0 * CHUNK, lds_base[0]);
    tdm_load_tile(xrow0 + 1 * CHUNK, lds_base[1]);

    float h = 0.0f;

    for (int c = 0; c < NCHUNK; ++c) {
        // TDM ops complete in order per wave: tile c done when <= 1 remains
        // in flight (tile c+1). Last chunk: wait for everything.
        if (c + 1 < NCHUNK) {
            asm volatile("s_wait_tensorcnt 0x1" ::: "memory");
        } else {
            asm volatile("s_wait_tensorcnt 0x0" ::: "memory");
        }

        const float* trow = &tile[c & 1][lane * LDS_STRIDE];

#pragma unroll 4
        for (int t = 0; t < CHUNK; ++t) {
            const float xv = trow[t];
            // pre-scaled input projection (F=1)
            const float xz = fmaf(xv, k0, bz);
            const float xr = fmaf(xv, k1, br);
            const float xh = fmaf(xv, k2, bh);
            // pre-scaled recurrent projection (UNITS=1)
            const float rz = fmaf(h, r0, cz);
            const float rr = fmaf(h, r1, cr);
            const float rh = fmaf(h, r2, ch);

            // z = sigmoid(.), r = sigmoid(.): v_exp_f32 + v_rcp_f32 each
            const float z  = __builtin_amdgcn_rcpf(1.0f + __builtin_amdgcn_exp2f(xz + rz));
            const float r  = __builtin_amdgcn_rcpf(1.0f + __builtin_amdgcn_exp2f(xr + rr));
            // tanh via e = exp2(pre-scaled arg)
            const float e  = __builtin_amdgcn_exp2f(xh + r * rh);
            const float hh = (1.0f - e) * __builtin_amdgcn_rcpf(1.0f + e);
            // hn = z*h + (1-z)*hh  ==  hh + z*(h - hh)
            const float hn = fmaf(z, h - hh, hh);
            h = (xv != 0.0f) ? hn : h;            // mask: x==0 -> keep h
        }

        // Refill the buffer we just consumed with tile c+2.
        if (c + 2 < NCHUNK) {
            // DS reads and TDM LDS writes use different counters and are
            // unordered: drain our LDS reads before the DMA overwrites them.
            asm volatile("s_wait_dscnt 0x0" ::: "memory");
            tdm_load_tile(xrow0 + (size_t)(c + 2) * CHUNK, lds_base[c & 1]);
        }
    }

    hout[row0 + lane] = h;
}

extern "C" void kernel_launch(void* const* d_in, const int* in_sizes, int n_in,
                              void* d_out, int out_size, void* d_ws, size_t ws_size,
                              hipStream_t stream) {
    (void)in_sizes; (void)n_in; (void)out_size; (void)d_ws; (void)ws_size;
    const float* x   = (const float*)d_in[0];   // (B, T, 1) fp32
    const float* k   = (const float*)d_in[1];   // (1, 3)
    const float* rk  = (const float*)d_in[2];   // (1, 3)
    const float* b   = (const float*)d_in[3];   // (2, 3)
    float* out = (float*)d_out;                 // (B, 1)

    gru_masked_tdm_kernel<<<B_DIM / ROWS, ROWS, 0, stream>>>(x, k, rk, b, out);
}